// RGCNEncoder_65687229825990
// MI455X (gfx1250) — compile-verified
//
#include <hip/hip_runtime.h>

// RGCN (block-diagonal) 2-layer encoder for MI455X / gfx1250.
//
// Strategy (see analysis): per-edge transform+scale scattered into an
// L2-resident (N,H) accumulator; dense root GEMM via V_WMMA_F32_16X16X4_F32.

typedef __attribute__((ext_vector_type(2))) float v2f;
typedef __attribute__((ext_vector_type(8))) float v8f;

#define HID  100   // hidden size
#define NBLK 5     // blocks
#define CB   20    // channels per block (HID / NBLK)
#define NREL 16    // relations
#define JT   7     // ceil(HID/16) output-column tiles for WMMA

// ---------------------------------------------------------------- zero fill
__global__ void rgcn_zero_kernel(float* __restrict__ p, long n) {
  long i = (long)blockIdx.x * blockDim.x + threadIdx.x;
  long stride = (long)gridDim.x * blockDim.x;
  for (; i < n; i += stride) p[i] = 0.0f;
}

// ------------------------------------------------- per-(relation,dst) counts
__global__ void rgcn_count_kernel(const int* __restrict__ dst,
                                  const int* __restrict__ et,
                                  float* __restrict__ cnt, int E, int N) {
  int e = blockIdx.x * blockDim.x + threadIdx.x;
  if (e < E) atomicAdd(&cnt[(long)et[e] * N + dst[e]], 1.0f);
}

// ----------------------------------------------------------- edge messages
// One thread per (edge, block): y = W[r][b] applied to x[src][b*20..],
// pre-scaled by 1/cnt[r,dst], atomically accumulated into agg[dst].
// agg is N*H = 40 MB -> L2 resident on MI455X (192 MB L2).
__global__ void rgcn_msg_kernel(const float* __restrict__ x,
                                const int* __restrict__ src,
                                const int* __restrict__ dst,
                                const int* __restrict__ et,
                                const float* __restrict__ W,
                                const float* __restrict__ cnt,
                                float* __restrict__ agg, int E, int N) {
  long t = (long)blockIdx.x * blockDim.x + threadIdx.x;
  if (t >= (long)E * NBLK) return;
  int e = (int)(t / NBLK);
  int b = (int)(t % NBLK);
  int s = src[e];
  int d = dst[e];
  int r = et[e];
  float inv = 1.0f / cnt[(long)r * N + d];  // cnt >= 1 for any existing edge

  const float* xr = x + (long)s * HID + b * CB;          // 16B aligned (80B mult)
  const float* w  = W + ((r * NBLK + b) * CB * CB);       // 20x20 block, row-major
  __builtin_prefetch(w, 0, 1);                            // global_prefetch_b8

  float xv[CB];
#pragma unroll
  for (int i = 0; i < CB; i += 4) {
    float4 v = *(const float4*)(xr + i);
    xv[i] = v.x; xv[i + 1] = v.y; xv[i + 2] = v.z; xv[i + 3] = v.w;
  }
  float acc[CB];
#pragma unroll
  for (int j = 0; j < CB; j++) acc[j] = 0.0f;

#pragma unroll 4
  for (int c = 0; c < CB; c++) {
    float xc = xv[c];
    const float4* wr = (const float4*)(w + c * CB);
#pragma unroll
    for (int q = 0; q < CB / 4; q++) {
      float4 wv = wr[q];
      acc[q * 4 + 0] += xc * wv.x;
      acc[q * 4 + 1] += xc * wv.y;
      acc[q * 4 + 2] += xc * wv.z;
      acc[q * 4 + 3] += xc * wv.w;
    }
  }
  float* ap = agg + (long)d * HID + b * CB;
#pragma unroll
  for (int j = 0; j < CB; j++) atomicAdd(ap + j, acc[j] * inv);
}

// --------------------------------------------- combine: agg + x@root + bias
// One wave32 per 16x16 output tile using V_WMMA_F32_16X16X4_F32.
// A (16x4 f32): lanes 0-15 hold M=0..15, VGPR pair = K=2*(lane/16)+{0,1}.
// B (4x16 f32): lanes 0-15 hold N=0..15, VGPR pair = K=2*(lane/16)+{0,1}.
// C/D (16x16 f32): VGPR i: lanes 0-15 -> M=i, lanes 16-31 -> M=i+8.
// Out-of-range rows/cols are CLAMPED (not zeroed): garbage only reaches
// C entries that are masked at writeback, so the K-loop is select-free
// and EXEC is all-1s at each WMMA (ISA requirement).
__global__ void rgcn_combine_kernel(const float* __restrict__ x,
                                    const float* __restrict__ root,
                                    const float* __restrict__ bias,
                                    const float* __restrict__ agg,
                                    float* __restrict__ out, int N, int doRelu) {
  int wavesPerBlock = blockDim.x >> 5;
  int wave = blockIdx.x * wavesPerBlock + (threadIdx.x >> 5);
  int lane = threadIdx.x & 31;
  int nTiles = ((N + 15) / 16) * JT;
  if (wave >= nTiles) return;  // wave-uniform exit

  int tj = wave % JT;
  int tn = wave / JT;
  int n0 = tn * 16, j0 = tj * 16;
  int lo = lane & 15, hi = lane >> 4;

  int an = n0 + lo; if (an > N - 1) an = N - 1;       // clamp (masked later)
  int bj = j0 + lo; if (bj > HID - 1) bj = HID - 1;   // clamp (masked later)
  const float* xp = x + (long)an * HID;

  v8f c = {0.f, 0.f, 0.f, 0.f, 0.f, 0.f, 0.f, 0.f};
#pragma unroll 5
  for (int k0 = 0; k0 < HID; k0 += 4) {               // 25 exact K steps
    int ka = k0 + 2 * hi;
    v2f a = { xp[ka], xp[ka + 1] };
    v2f b = { root[ka * HID + bj], root[(ka + 1) * HID + bj] };
    c = __builtin_amdgcn_wmma_f32_16x16x4_f32(
        /*neg_a=*/false, a, /*neg_b=*/false, b,
        /*c_mod=*/(short)0, c, /*reuse_a=*/false, /*reuse_b=*/false);
  }

  float bv = bias[bj];
#pragma unroll
  for (int i = 0; i < 8; i++) {
    int row = n0 + i + 8 * hi;
    int col = j0 + lo;
    if (row < N && col < HID) {
      float v = c[i] + agg[(long)row * HID + col] + bv;
      if (doRelu) v = v > 0.0f ? v : 0.0f;
      out[(long)row * HID + col] = v;
    }
  }
}

// ------------------------------------------------------------------ driver
extern "C" void kernel_launch(void* const* d_in, const int* in_sizes, int n_in,
                              void* d_out, int out_size, void* d_ws, size_t ws_size,
                              hipStream_t stream) {
  const float* node = (const float*)d_in[0];
  const float* w1   = (const float*)d_in[1];
  const float* r1   = (const float*)d_in[2];
  const float* b1   = (const float*)d_in[3];
  const float* w2   = (const float*)d_in[4];
  const float* r2   = (const float*)d_in[5];
  const float* b2   = (const float*)d_in[6];
  const int*   eidx = (const int*)d_in[7];   // (2, E): [src row | dst row]
  const int*   et   = (const int*)d_in[8];   // (E,)
  int E = in_sizes[8];
  int N = in_sizes[0] / HID;
  const int* src = eidx;
  const int* dst = eidx + E;
  float* out = (float*)d_out;

  // Workspace layout: [agg: N*H][cnt: R*N][x1: N*H]  (~86.4 MB total)
  float* agg = (float*)d_ws;
  float* cnt = agg + (long)N * HID;
  float* x1  = cnt + (long)NREL * N;

  int tiles = ((N + 15) / 16) * JT;
  unsigned msgBlocks   = (unsigned)(((long)E * NBLK + 255) / 256);
  unsigned cntBlocks   = (unsigned)((E + 255) / 256);
  unsigned tileBlocks  = (unsigned)((tiles + 7) / 8);   // 8 waves / 256-thread block

  // ---- counts (edge-only; shared by both layers) + zero agg
  rgcn_zero_kernel<<<2048, 256, 0, stream>>>(agg, (long)N * HID + (long)NREL * N);
  rgcn_count_kernel<<<cntBlocks, 256, 0, stream>>>(dst, et, cnt, E, N);

  // ---- layer 1
  rgcn_msg_kernel<<<msgBlocks, 256, 0, stream>>>(node, src, dst, et, w1, cnt, agg, E, N);
  rgcn_combine_kernel<<<tileBlocks, 256, 0, stream>>>(node, r1, b1, agg, x1, N, /*relu=*/1);

  // ---- layer 2
  rgcn_zero_kernel<<<2048, 256, 0, stream>>>(agg, (long)N * HID);
  rgcn_msg_kernel<<<msgBlocks, 256, 0, stream>>>(x1, src, dst, et, w2, cnt, agg, E, N);
  rgcn_combine_kernel<<<tileBlocks, 256, 0, stream>>>(x1, r2, b2, agg, out, N, /*relu=*/0);
}